// PatchAttention_850403525314
// MI455X (gfx1250) — compile-verified
//
#include <hip/hip_runtime.h>
#include <hip/hip_bf16.h>

// ---- problem constants ----
#define BB   2
#define NN   256
#define DM   512
#define HH   8
#define HD   64
#define QKVW (3*DM)   // 1536

typedef float v2f __attribute__((ext_vector_type(2)));
typedef float v8f __attribute__((ext_vector_type(8)));

static __device__ __forceinline__ v8f zero8() {
  v8f z;
  #pragma unroll
  for (int i = 0; i < 8; ++i) z[i] = 0.0f;
  return z;
}

// D = A(16x4) * B(4x16) + C, f32, wave32.
static __device__ __forceinline__ v8f wmma4(v2f a, v2f b, v8f c) {
  return __builtin_amdgcn_wmma_f32_16x16x4_f32(false, a, false, b, (short)0, c,
                                               false, false);
}

static __device__ __forceinline__ float silu(float x) {
  return x / (1.0f + __expf(-x));
}

// CDNA5 async copy global -> LDS (ASYNCcnt-tracked, bypasses VGPRs)
static __device__ __forceinline__ void async_g2l_b128(unsigned lds_off, const void* gaddr) {
  asm volatile("global_load_async_to_lds_b128 %0, %1, off"
               :: "v"(lds_off), "v"(gaddr) : "memory");
}
static __device__ __forceinline__ void wait_async0() {
  asm volatile("s_wait_asynccnt 0" ::: "memory");
}

// =====================================================================
// K1: qkv = x @ Wqkv^T + bqkv    (512 rows x 512 K x 1536 cols)
// =====================================================================
__global__ void k_qkv(const float* __restrict__ x, const float* __restrict__ Wqkv,
                      const float* __restrict__ bqkv, float* __restrict__ qkv) {
  const int lane = threadIdx.x;
  const int lm = lane & 15, half = lane >> 4;
  const int rowbase = blockIdx.x * 16;
  const int colbase = blockIdx.y * 64;
  v8f acc[4];
  #pragma unroll
  for (int t = 0; t < 4; ++t) acc[t] = zero8();

  for (int kk = 0; kk < DM / 4; ++kk) {
    const int kb = kk * 4 + 2 * half;
    const v2f a = *(const v2f*)&x[(rowbase + lm) * DM + kb];
    #pragma unroll
    for (int t = 0; t < 4; ++t) {
      const int col = colbase + t * 16 + lm;
      const v2f b = *(const v2f*)&Wqkv[col * DM + kb];
      acc[t] = wmma4(a, b, acc[t]);
    }
  }
  #pragma unroll
  for (int t = 0; t < 4; ++t) {
    const int col = colbase + t * 16 + lm;
    const float bias = bqkv[col];
    #pragma unroll
    for (int r = 0; r < 8; ++r) {
      const int row = rowbase + r + 8 * half;
      qkv[row * QKVW + col] = acc[t][r] + bias;
    }
  }
}

// =====================================================================
// K2: rel_bias[b,h,i,j] = silu(rel @ Wb1p) @ Wb2^T + bb2  (fused MLP)
// =====================================================================
__global__ void k_relbias(const float* __restrict__ coords, const float* __restrict__ Wb1,
                          const float* __restrict__ bb1, const float* __restrict__ Wb2,
                          const float* __restrict__ bb2, float* __restrict__ attn) {
  __shared__ __align__(16) float hblds[4][16 * 17];
  const int tid = threadIdx.x;
  const int lane = tid & 31, wave = tid >> 5;
  const int lm = lane & 15, half = lane >> 4;
  const int i = blockIdx.x;
  const int bb = blockIdx.y;

  const float cx = coords[(bb * NN + i) * 3 + 0];
  const float cy = coords[(bb * NN + i) * 3 + 1];
  const float cz = coords[(bb * NN + i) * 3 + 2];

  v2f relA[4];
  #pragma unroll
  for (int q = 0; q < 4; ++q) {
    const int j = (wave * 4 + q) * 16 + lm;
    const float r0 = cx - coords[(bb * NN + j) * 3 + 0];
    const float r1 = cy - coords[(bb * NN + j) * 3 + 1];
    const float r2 = cz - coords[(bb * NN + j) * 3 + 2];
    relA[q].x = half ? r2 : r0;
    relA[q].y = half ? 1.0f : r1;
  }

  v8f acc[4];
  #pragma unroll
  for (int q = 0; q < 4; ++q) acc[q] = zero8();
  float* my = hblds[wave];

  for (int ct = 0; ct < 32; ++ct) {
    const int c = ct * 16 + lm;
    v2f bW;                       // Wb1 padded: rows {w0,w1,w2,bias}
    if (half == 0) { bW.x = Wb1[c * 3 + 0]; bW.y = Wb1[c * 3 + 1]; }
    else           { bW.x = Wb1[c * 3 + 2]; bW.y = bb1[c]; }

    v2f b2[4];                    // Wb2^T chunks (K=c, N=h)
    #pragma unroll
    for (int k = 0; k < 4; ++k) {
      const int kb = ct * 16 + k * 4 + 2 * half;
      b2[k].x = (lm < HH) ? Wb2[lm * DM + kb] : 0.0f;
      b2[k].y = (lm < HH) ? Wb2[lm * DM + kb + 1] : 0.0f;
    }

    #pragma unroll
    for (int q = 0; q < 4; ++q) {
      v8f hb = wmma4(relA[q], bW, zero8());
      #pragma unroll
      for (int r = 0; r < 8; ++r) hb[r] = silu(hb[r]);
      #pragma unroll
      for (int r = 0; r < 8; ++r) my[(r + 8 * half) * 17 + lm] = hb[r];
      #pragma unroll
      for (int k = 0; k < 4; ++k) {
        const int k0 = k * 4 + 2 * half;
        v2f aH;
        aH.x = my[lm * 17 + k0];
        aH.y = my[lm * 17 + k0 + 1];
        acc[q] = wmma4(aH, b2[k], acc[q]);
      }
    }
  }
  const float bias2 = (lm < HH) ? bb2[lm] : 0.0f;
  #pragma unroll
  for (int q = 0; q < 4; ++q) {
    if (lm < HH) {
      #pragma unroll
      for (int r = 0; r < 8; ++r) {
        const int j = (wave * 4 + q) * 16 + r + 8 * half;
        attn[((bb * HH + lm) * NN + i) * NN + j] = acc[q][r] + bias2;
      }
    }
  }
}

// =====================================================================
// K3: scores = q@k^T*scale + bias -> softmax -> weights (in-place)
//     plus fused base = weights @ v  -> basews
// =====================================================================
__global__ void k_attn(const float* __restrict__ qkv, float* __restrict__ attn,
                       float* __restrict__ basews) {
  __shared__ __align__(16) float S[16 * 257];
  __shared__ float red[16][8];
  const int tid = threadIdx.x;
  const int lane = tid & 31, wave = tid >> 5;
  const int lm = lane & 15, half = lane >> 4;
  const int it = blockIdx.x, h = blockIdx.y, bb = blockIdx.z;

  // ---- Step A: scores tile 16 x 256 ----
  const int qrow = bb * NN + it * 16 + lm;
  v2f aq[16];
  #pragma unroll
  for (int t = 0; t < 16; ++t) {
    const int kb = t * 4 + 2 * half;
    aq[t] = *(const v2f*)&qkv[qrow * QKVW + h * HD + kb];
  }
  for (int q = 0; q < 4; ++q) {
    const int jt = wave * 4 + q;
    const int j = jt * 16 + lm;
    v8f acc = zero8();
    #pragma unroll
    for (int t = 0; t < 16; ++t) {
      const int kb = t * 4 + 2 * half;
      v2f bk = *(const v2f*)&qkv[(bb * NN + j) * QKVW + DM + h * HD + kb];
      acc = wmma4(aq[t], bk, acc);
    }
    #pragma unroll
    for (int r = 0; r < 8; ++r) {
      const int m = r + 8 * half;
      const float bias = attn[((bb * HH + h) * NN + it * 16 + m) * NN + jt * 16 + lm];
      S[m * 257 + jt * 16 + lm] = acc[r] * 0.125f + bias;
    }
  }
  __syncthreads();

  // ---- Step B: row softmax (16 rows x 8 threads each) ----
  const int row = tid >> 3, sub = tid & 7;
  float mx = -3.4e38f;
  for (int k = 0; k < 32; ++k) mx = fmaxf(mx, S[row * 257 + k * 8 + sub]);
  red[row][sub] = mx;
  __syncthreads();
  float rmx = red[row][0];
  #pragma unroll
  for (int k = 1; k < 8; ++k) rmx = fmaxf(rmx, red[row][k]);
  float sum = 0.0f;
  for (int k = 0; k < 32; ++k) {
    const int c = k * 8 + sub;
    const float e = __expf(S[row * 257 + c] - rmx);
    S[row * 257 + c] = e;
    sum += e;
  }
  __syncthreads();
  red[row][sub] = sum;
  __syncthreads();
  float rsum = red[row][0];
  #pragma unroll
  for (int k = 1; k < 8; ++k) rsum += red[row][k];
  const float inv = 1.0f / rsum;
  for (int k = 0; k < 32; ++k) {
    const int c = k * 8 + sub;
    const float w = S[row * 257 + c] * inv;
    S[row * 257 + c] = w;
    attn[((bb * HH + h) * NN + it * 16 + row) * NN + c] = w;
  }
  __syncthreads();

  // ---- Step C: base = W(16x256) @ V(256x64); wave -> d-tile ----
  const int d = wave * 16 + lm;
  v8f acc = zero8();
  for (int t = 0; t < 64; ++t) {
    const int k0 = t * 4 + 2 * half;
    v2f aw;
    aw.x = S[lm * 257 + k0];
    aw.y = S[lm * 257 + k0 + 1];
    v2f bv;
    bv.x = qkv[(bb * NN + k0) * QKVW + 2 * DM + h * HD + d];
    bv.y = qkv[(bb * NN + k0 + 1) * QKVW + 2 * DM + h * HD + d];
    acc = wmma4(aw, bv, acc);
  }
  #pragma unroll
  for (int r = 0; r < 8; ++r) {
    const int m = r + 8 * half;
    basews[((bb * HH + h) * NN + it * 16 + m) * HD + d] = acc[r];
  }
}

// =====================================================================
// K4: G[b,i,h,c] = sum_j weights[b,h,i,j] * silu(rel@Wv1p)[j,c]
// async-LDS staging of weights + coords; jt-outer loop with 8 live
// G accumulators; double-buffered hv tile for wmma1/wmma2 overlap
// =====================================================================
__global__ void k_gmat(const float* __restrict__ coords, const float* __restrict__ Wv1,
                       const float* __restrict__ bv1, const float* __restrict__ attn,
                       float* __restrict__ G) {
  __shared__ __align__(16) float wlds[HH * NN];        // 8 x 256 weights
  __shared__ __align__(16) float clds[NN * 3];         // coords for this batch
  __shared__ __align__(16) float hvlds[4][2][16 * 17]; // double-buffered hv tiles
  const int tid = threadIdx.x;
  const int lane = tid & 31, wave = tid >> 5;
  const int lm = lane & 15, half = lane >> 4;
  const int i = blockIdx.x, bb = blockIdx.y;

  // ---- async global->LDS staging (TDM-class path, ASYNCcnt) ----
  #pragma unroll
  for (int k = 0; k < 4; ++k) {
    const int e4 = tid + k * 128;          // 512 chunks of 16B = 8KB weights
    const int hh = e4 >> 6;
    const int j4 = (e4 & 63) * 4;
    const unsigned loff = (unsigned)(unsigned long long)&wlds[hh * NN + j4];
    async_g2l_b128(loff, (const void*)&attn[((bb * HH + hh) * NN + i) * NN + j4]);
  }
  #pragma unroll
  for (int k = 0; k < 2; ++k) {
    const int e = tid + k * 128;           // 192 chunks of 16B = 3KB coords
    if (e < (NN * 3) / 4) {
      const unsigned loff = (unsigned)(unsigned long long)&clds[e * 4];
      async_g2l_b128(loff, (const void*)&coords[bb * NN * 3 + e * 4]);
    }
  }
  wait_async0();
  __syncthreads();

  const float cx = clds[i * 3 + 0];
  const float cy = clds[i * 3 + 1];
  const float cz = clds[i * 3 + 2];
  v2f relA[16];
  #pragma unroll
  for (int jt = 0; jt < 16; ++jt) {
    const int j = jt * 16 + lm;
    const float r0 = cx - clds[j * 3 + 0];
    const float r1 = cy - clds[j * 3 + 1];
    const float r2 = cz - clds[j * 3 + 2];
    relA[jt].x = half ? r2 : r0;
    relA[jt].y = half ? 1.0f : r1;
  }

  // Wv1 padded B operands for this wave's 8 c-tiles (hoisted)
  v2f bW[8];
  #pragma unroll
  for (int cc = 0; cc < 8; ++cc) {
    const int c = (wave * 8 + cc) * 16 + lm;
    if (half == 0) { bW[cc].x = Wv1[c * 3 + 0]; bW[cc].y = Wv1[c * 3 + 1]; }
    else           { bW[cc].x = Wv1[c * 3 + 2]; bW[cc].y = bv1[c]; }
  }

  v8f gacc[8];
  #pragma unroll
  for (int cc = 0; cc < 8; ++cc) gacc[cc] = zero8();

  for (int jt = 0; jt < 16; ++jt) {
    // weight A-chunks for this j-tile (read once, reused across 8 c-tiles)
    v2f aw[4];
    #pragma unroll
    for (int k = 0; k < 4; ++k) {
      const int kA = jt * 16 + k * 4 + 2 * half;
      aw[k].x = (lm < HH) ? wlds[lm * NN + kA] : 0.0f;
      aw[k].y = (lm < HH) ? wlds[lm * NN + kA + 1] : 0.0f;
    }
    #pragma unroll
    for (int cc = 0; cc < 8; ++cc) {
      v8f hv = wmma4(relA[jt], bW[cc], zero8());
      #pragma unroll
      for (int r = 0; r < 8; ++r) hv[r] = silu(hv[r]);
      float* my = hvlds[wave][cc & 1];
      #pragma unroll
      for (int r = 0; r < 8; ++r) my[(r + 8 * half) * 17 + lm] = hv[r];
      #pragma unroll
      for (int k = 0; k < 4; ++k) {
        const int k0 = k * 4 + 2 * half;
        v2f bh;
        bh.x = my[k0 * 17 + lm];
        bh.y = my[(k0 + 1) * 17 + lm];
        gacc[cc] = wmma4(aw[k], bh, gacc[cc]);
      }
    }
  }
  if (half == 0) {   // D rows M=h: only h<8 valid
    #pragma unroll
    for (int cc = 0; cc < 8; ++cc) {
      const int c = (wave * 8 + cc) * 16 + lm;
      #pragma unroll
      for (int r = 0; r < 8; ++r)
        G[((bb * NN + i) * HH + r) * DM + c] = gacc[cc][r];
    }
  }
}

// =====================================================================
// K5: out[b,i,h*64+d] = G[b,i,h,:] @ Wv2[h*64+d,:] + bv2 + base
// =====================================================================
__global__ void k_out(const float* __restrict__ G, const float* __restrict__ Wv2,
                      const float* __restrict__ bv2, const float* __restrict__ basews,
                      float* __restrict__ out) {
  const int tid = threadIdx.x;
  const int lane = tid & 31, wave = tid >> 5;
  const int lm = lane & 15, half = lane >> 4;
  const int it = blockIdx.x, h = blockIdx.y, bb = blockIdx.z;

  const int d = wave * 16 + lm;
  const int grow = ((bb * NN + it * 16 + lm) * HH + h) * DM;
  const int wrow = (h * HD + d) * DM;
  __builtin_prefetch(&Wv2[wrow], 0, 3);

  v8f acc = zero8();
  for (int t = 0; t < 128; ++t) {
    const int k0 = t * 4 + 2 * half;
    const v2f a = *(const v2f*)&G[grow + k0];
    const v2f b = *(const v2f*)&Wv2[wrow + k0];
    acc = wmma4(a, b, acc);
  }
  const float bias = bv2[h * HD + d];
  #pragma unroll
  for (int r = 0; r < 8; ++r) {
    const int m = r + 8 * half;
    out[(bb * NN + it * 16 + m) * DM + h * HD + d] =
        acc[r] + bias + basews[((bb * HH + h) * NN + it * 16 + m) * HD + d];
  }
}

// =====================================================================
extern "C" void kernel_launch(void* const* d_in, const int* in_sizes, int n_in,
                              void* d_out, int out_size, void* d_ws, size_t ws_size,
                              hipStream_t stream) {
  (void)in_sizes; (void)n_in; (void)out_size; (void)ws_size;
  const float* x      = (const float*)d_in[0];
  const float* coords = (const float*)d_in[1];
  const float* Wqkv   = (const float*)d_in[2];
  const float* bqkv   = (const float*)d_in[3];
  const float* Wb1    = (const float*)d_in[4];
  const float* bb1    = (const float*)d_in[5];
  const float* Wb2    = (const float*)d_in[6];
  const float* bb2    = (const float*)d_in[7];
  const float* Wv1    = (const float*)d_in[8];
  const float* bv1    = (const float*)d_in[9];
  const float* Wv2    = (const float*)d_in[10];
  const float* bv2    = (const float*)d_in[11];

  float* ws   = (float*)d_ws;
  float* qkv  = ws;                         // B*N*1536      = 786432
  float* attn = qkv + (BB * NN * QKVW);     // B*H*N*N       = 1048576
  float* base = attn + (BB * HH * NN * NN); // B*H*N*HD      = 262144
  float* G    = base + (BB * HH * NN * HD); // B*N*H*512     = 2097152
  // total: 16 MB of f32 workspace (L2-resident on MI455X)

  k_qkv    <<<dim3(32, 24),   32,  0, stream>>>(x, Wqkv, bqkv, qkv);
  k_relbias<<<dim3(256, 2),   128, 0, stream>>>(coords, Wb1, bb1, Wb2, bb2, attn);
  k_attn   <<<dim3(16, 8, 2), 128, 0, stream>>>(qkv, attn, base);
  k_gmat   <<<dim3(256, 2),   128, 0, stream>>>(coords, Wv1, bv1, attn, G);
  k_out    <<<dim3(16, 8, 2), 128, 0, stream>>>(G, Wv2, bv2, base, (float*)d_out);
}